// MixLogCDFCoupling_24180665876837
// MI455X (gfx1250) — compile-verified
//
#include <hip/hip_runtime.h>
#include <math.h>

// MixLogCDF coupling, MI455X (gfx1250, wave32).
// Bandwidth-bound streaming kernel: ~434 MB traffic -> ~19 us at 23.3 TB/s.
// pi cached in VGPRs, mu/s streamed once with online logsumexp, NT cache hints,
// deterministic two-stage sldj reduction using V_WMMA_F32_16X16X4_F32 for the
// 32-lane wave reduction (matrix pipe, full f32 precision).

typedef __attribute__((ext_vector_type(2))) float v2f;
typedef __attribute__((ext_vector_type(8))) float v8f;

#define TPB 256

// Sum 32 lane values with one f32 WMMA (16x16x4): A holds the 32 partials
// (A[m,0]=p_m from lanes 0-15, A[m,2]=p_{m+16} from lanes 16-31, other K=0),
// B = ones -> D[m,n] = p_m + p_{m+16}. Summing the 8 D VGPRs per lane gives
// rows 0-7 (lanes<16) / rows 8-15 (lanes>=16); one xor-16 shuffle combines.
// Requires EXEC == all ones (guaranteed at call sites below).
__device__ __forceinline__ float wave_sum_wmma(float p) {
  v2f A;  A[0] = p;    A[1] = 0.0f;
  v2f Bo; Bo[0] = 1.0f; Bo[1] = 1.0f;
  v8f C = {0.f, 0.f, 0.f, 0.f, 0.f, 0.f, 0.f, 0.f};
  v8f D = __builtin_amdgcn_wmma_f32_16x16x4_f32(false, A, false, Bo,
                                                (short)0, C, false, false);
  float t = ((D[0] + D[1]) + (D[2] + D[3])) + ((D[4] + D[5]) + (D[6] + D[7]));
  t += __shfl_xor(t, 16, 32);
  return t;
}

// Block-wide sum; result valid in thread 0.
__device__ __forceinline__ float block_sum(float p, float* smem) {
  const float w = wave_sum_wmma(p);
  const int wid = threadIdx.x >> 5;
  if ((threadIdx.x & 31) == 0) smem[wid] = w;
  __syncthreads();
  float tot = 0.0f;
  if (threadIdx.x == 0) {
#pragma unroll
    for (int i = 0; i < TPB / 32; ++i) tot += smem[i];
  }
  return tot;
}

__global__ void __launch_bounds__(TPB)
mixlogcdf_main(const float* __restrict__ x, const float* __restrict__ a,
               const float* __restrict__ b, const float* __restrict__ pi,
               const float* __restrict__ mu, const float* __restrict__ s,
               float* __restrict__ y, float* __restrict__ partials,
               int ChHW, int CHW, int blocksPerBatch, int n) {
  constexpr int K = 16;
  __shared__ float smem[TPB / 32];

  const int bidx = blockIdx.x / blocksPerBatch;   // uniform per block
  const int idx  = blockIdx.x * TPB + threadIdx.x;

  float contrib = 0.0f;
  if (idx < n) {
    const int    rem   = idx - bidx * ChHW;       // c*HW + h*W + w, c < Ch
    const size_t xbase = (size_t)bidx * CHW + rem;
    const float  xv    = __builtin_nontemporal_load(&x[xbase]);

    const size_t mixbase = (size_t)bidx * (size_t)K * ChHW + rem;
    const float* __restrict__ pib = pi + mixbase;
    const float* __restrict__ mub = mu + mixbase;
    const float* __restrict__ sb  = s  + mixbase;

    // Pass 1: pi into registers (reused twice), running max.
    float pir[K];
    float pmax = -INFINITY;
#pragma unroll
    for (int k = 0; k < K; ++k) {
      pir[k] = __builtin_nontemporal_load(&pib[(size_t)k * ChHW]);
      pmax = fmaxf(pmax, pir[k]);
    }
    float psum = 0.0f;
#pragma unroll
    for (int k = 0; k < K; ++k) psum += __expf(pir[k] - pmax);
    const float lse_pi = pmax + __logf(psum);     // logsumexp(pi) over K

    // Pass 2: stream mu/s once; online logsumexp for log_cdf and log_pdf.
    float m1 = -INFINITY, s1 = 0.0f;              // CDF accumulator
    float m2 = -INFINITY, s2 = 0.0f;              // PDF accumulator
#pragma unroll
    for (int k = 0; k < K; ++k) {
      const float muk = __builtin_nontemporal_load(&mub[(size_t)k * ChHW]);
      const float sk  = __builtin_nontemporal_load(&sb [(size_t)k * ChHW]);
      const float z   = (xv - muk) * __expf(-sk);
      // softplus(-z) = max(-z,0) + log1p(exp(-|z|))   (stable)
      const float sp  = fmaxf(-z, 0.0f) + log1pf(__expf(-fabsf(z)));
      const float w   = pir[k] - lse_pi;           // log_softmax(pi)
      const float t1  = w - sp;                    // + log_sigmoid(z)
      const float t2  = w - z - sk - 2.0f * sp;    // + logistic log_pdf

      float nm = fmaxf(m1, t1);
      s1 = s1 * __expf(m1 - nm) + __expf(t1 - nm);
      m1 = nm;
      nm = fmaxf(m2, t2);
      s2 = s2 * __expf(m2 - nm) + __expf(t2 - nm);
      m2 = nm;
    }
    const float log_cdf      = m1 + __logf(s1);
    const float logistic_ldj = m2 + __logf(s2);

    const float u    = __expf(log_cdf);
    const float l1mu = log1pf(-u);                 // log1p(-u)
    const float av   = __builtin_nontemporal_load(&a[idx]);
    const float bv   = __builtin_nontemporal_load(&b[idx]);
    const float outv = (log_cdf - l1mu + bv) * __expf(av);

    __builtin_nontemporal_store(outv, &y[xbase]);
    // identity half passthrough: x[b, Ch + c, h, w]
    const float xid = __builtin_nontemporal_load(&x[xbase + ChHW]);
    __builtin_nontemporal_store(xid, &y[xbase + ChHW]);

    // scale_ldj = -log(u) - log1p(-u)
    contrib = logistic_ldj + (-log_cdf - l1mu) + av;
  }

  // Control flow has reconverged; EXEC is all-ones here (n % TPB == 0).
  const float tot = block_sum(contrib, smem);
  if (threadIdx.x == 0) partials[blockIdx.x] = tot;
}

__global__ void __launch_bounds__(TPB)
mixlogcdf_finalize(const float* __restrict__ partials,
                   const float* __restrict__ sldj_in,
                   float* __restrict__ sldj_out, int blocksPerBatch) {
  __shared__ float smem[TPB / 32];
  const int bb = blockIdx.x;                       // one block per batch
  float p = 0.0f;
  for (int i = threadIdx.x; i < blocksPerBatch; i += TPB)
    p += partials[(size_t)bb * blocksPerBatch + i];
  const float tot = block_sum(p, smem);            // deterministic order
  if (threadIdx.x == 0) sldj_out[bb] = sldj_in[bb] + tot;
}

extern "C" void kernel_launch(void* const* d_in, const int* in_sizes, int n_in,
                              void* d_out, int out_size, void* d_ws, size_t ws_size,
                              hipStream_t stream) {
  const float* x    = (const float*)d_in[0];
  const float* sldj = (const float*)d_in[1];
  const float* a    = (const float*)d_in[2];
  const float* b    = (const float*)d_in[3];
  const float* pi   = (const float*)d_in[4];
  const float* mu   = (const float*)d_in[5];
  const float* s    = (const float*)d_in[6];

  const int nX = in_sizes[0];          // B*C*H*W
  const int Bb = in_sizes[1];          // B
  const int n  = in_sizes[2];          // B*Ch*H*W
  const int ChHW = n / Bb;             // Ch*H*W (131072, multiple of TPB)
  const int CHW  = nX / Bb;            // C*H*W

  const int blocks = (n + TPB - 1) / TPB;
  const int blocksPerBatch = blocks / Bb;

  float* y        = (float*)d_out;     // (B, C, H, W) flat
  float* sldj_out = y + nX;            // (B,) appended
  float* partials = (float*)d_ws;      // blocks floats (32 KB)

  mixlogcdf_main<<<blocks, TPB, 0, stream>>>(x, a, b, pi, mu, s, y, partials,
                                             ChHW, CHW, blocksPerBatch, n);
  mixlogcdf_finalize<<<Bb, TPB, 0, stream>>>(partials, sldj, sldj_out,
                                             blocksPerBatch);
}